// Classifier_23545010717370
// MI455X (gfx1250) — compile-verified
//
#include <hip/hip_runtime.h>
#include <cstdint>

typedef float v2f __attribute__((ext_vector_type(2)));
typedef float v8f __attribute__((ext_vector_type(8)));

namespace {
constexpr int   kB      = 64;
constexpr int   kN      = 2048;
constexpr int   kD      = 128;
constexpr int   kChunk  = 32;              // tokens per LDS buffer
constexpr int   kNChunk = kN / kChunk;     // 64
constexpr float kCurv   = 0.1f;
constexpr float kEps    = 1e-15f;
}

// Async DMA: each lane copies 16B global -> LDS; tracked by ASYNCcnt.
__device__ __forceinline__ void async_b128(uint32_t lds_addr, const void* gaddr) {
  asm volatile("global_load_async_to_lds_b128 %0, %1, off"
               :: "v"(lds_addr), "v"(gaddr)
               : "memory");
}

// VALU-only lanes 0-15 <-> 16-31 exchange (identity selects), no DS pipe.
__device__ __forceinline__ float xor16_add(float s) {
  unsigned si = __builtin_bit_cast(unsigned, s);
  unsigned sw = __builtin_amdgcn_permlanex16(si, si, 0x76543210u, 0xfedcba98u,
                                             false, false);
  return s + __builtin_bit_cast(float, sw);
}

// 32-lane allreduce-add using the matrix pipe.
// A[m,0..3]: lane m<16 holds {p_m, 0}, lane m+16 holds {p_(m+16), 0}.
// B = all-ones (layout-independent) => C[m,n] = p_m + p_(m+16) broadcast over n.
// In-lane tree sum over the 8 C VGPRs gives half-sums; one permlanex16
// exchange completes and broadcasts the 32-lane total.
__device__ __forceinline__ float allreduce32(float p) {
  v2f a;    a[0] = p;    a[1] = 0.0f;
  v2f ones; ones[0] = 1.0f; ones[1] = 1.0f;
  v8f c = {0.0f, 0.0f, 0.0f, 0.0f, 0.0f, 0.0f, 0.0f, 0.0f};
  c = __builtin_amdgcn_wmma_f32_16x16x4_f32(false, a, false, ones,
                                            (short)0, c, false, false);
  float s = ((c[0] + c[1]) + (c[2] + c[3])) + ((c[4] + c[5]) + (c[6] + c[7]));
  return xor16_add(s);
}

__global__ __launch_bounds__(32) void hyp_fold_kernel(
    const float* __restrict__ x, const int* __restrict__ mask,
    float* __restrict__ out) {
  __shared__ __align__(16) float shx[2][kChunk * kD];  // 32 KB double buffer
  __shared__ __align__(16) int   shm[kN];              // 8 KB mask row

  const int b = blockIdx.x;
  const int t = threadIdx.x;  // 0..31, one wave per batch row

  const float* xrow = x + (size_t)b * kN * kD;
  const char*  mrow = (const char*)(mask + (size_t)b * kN);

  // One-time async prefetch of the whole mask row: 8192 B = 16 x (32 lanes x 16B).
  {
    const uint32_t l0 = (uint32_t)(uintptr_t)&shm[0];
#pragma unroll
    for (int j = 0; j < 16; ++j)
      async_b128(l0 + j * 512 + t * 16, mrow + j * 512 + t * 16);
  }
  // Prefetch x chunk 0: kChunk*kD*4 = 16384 B = 32 async b128 ops.
  {
    const uint32_t l0 = (uint32_t)(uintptr_t)&shx[0][0];
    const char*    g0 = (const char*)xrow;
#pragma unroll
    for (int j = 0; j < 32; ++j)
      async_b128(l0 + j * 512 + t * 16, g0 + j * 512 + t * 16);
  }

  float4 acc = make_float4(0.0f, 0.0f, 0.0f, 0.0f);
  int cnt = 0;
  int buf = 0;

  for (int c = 0; c < kNChunk; ++c) {
    if (c + 1 < kNChunk) {
      // Issue next chunk first, then wait for the current one (cnt <= 32).
      const uint32_t l0 = (uint32_t)(uintptr_t)&shx[buf ^ 1][0];
      const char*    g0 = (const char*)(xrow + (size_t)(c + 1) * kChunk * kD);
#pragma unroll
      for (int j = 0; j < 32; ++j)
        async_b128(l0 + j * 512 + t * 16, g0 + j * 512 + t * 16);
      asm volatile("s_wait_asynccnt 0x20" ::: "memory");
    } else {
      asm volatile("s_wait_asynccnt 0x0" ::: "memory");
    }

    const float* xb = &shx[buf][0];
    // Whole chunk's masks in one DS access: lane i holds mask[base+i].
    const int mv = shm[c * kChunk + t];

#pragma unroll 4
    for (int i = 0; i < kChunk; ++i) {
      // Uniform per-step mask straight from an SGPR (v_readlane), no DS op.
      const int m = __builtin_amdgcn_readlane(mv, i);
      const float4 xi = *(const float4*)(xb + i * kD + t * 4);

      float pxy = acc.x * xi.x + acc.y * xi.y + acc.z * xi.z + acc.w * xi.w;
      float py2 = xi.x * xi.x + xi.y * xi.y + xi.z * xi.z + xi.w * xi.w;
      float px2 = acc.x * acc.x + acc.y * acc.y + acc.z * acc.z + acc.w * acc.w;
      // Pin the load + cheap partials outside the mask branch so ds_loads keep
      // issuing ahead; only the expensive reduce+update is skipped when m==0.
      asm("" : "+v"(pxy), "+v"(py2), "+v"(px2));

      if (m != 0) {
        const float xy = allreduce32(pxy);
        const float y2 = allreduce32(py2);
        const float x2 = allreduce32(px2);

        const float two_cxy = 2.0f * kCurv * xy;
        const float ca  = 1.0f + two_cxy + kCurv * y2;
        const float cb  = 1.0f - kCurv * x2;
        const float den = fmaxf(1.0f + two_cxy + kCurv * kCurv * x2 * y2, kEps);
        const float inv = __builtin_amdgcn_rcpf(den);

        acc.x = (ca * acc.x + cb * xi.x) * inv;
        acc.y = (ca * acc.y + cb * xi.y) * inv;
        acc.z = (ca * acc.z + cb * xi.z) * inv;
        acc.w = (ca * acc.w + cb * xi.w) * inv;
        ++cnt;
      }
    }
    buf ^= 1;
  }

  // Mobius scalar-mul tail: r = 1/max(cnt,1).
  float pn = acc.x * acc.x + acc.y * acc.y + acc.z * acc.z + acc.w * acc.w;
  const float n2   = allreduce32(pn);
  const float norm = sqrtf(fmaxf(n2, kEps));
  const float sc   = 0.316227766016837933f;  // sqrt(0.1)
  const float arg  = fminf(fmaxf(sc * norm, 0.0f), 1.0f - 1e-7f);
  const float r    = 1.0f / (float)(cnt > 0 ? cnt : 1);
  const float th   = tanhf(r * atanhf(arg));
  const float scale = th / (sc * norm);

  float4 o = make_float4(acc.x * scale, acc.y * scale,
                         acc.z * scale, acc.w * scale);
  const float4 x0 = *(const float4*)(xrow + t * 4);  // unconditional, uniform select
  if (cnt == 0) o = x0;
  *(float4*)(out + (size_t)b * kD + t * 4) = o;
}

extern "C" void kernel_launch(void* const* d_in, const int* in_sizes, int n_in,
                              void* d_out, int out_size, void* d_ws, size_t ws_size,
                              hipStream_t stream) {
  (void)in_sizes; (void)n_in; (void)out_size; (void)d_ws; (void)ws_size;
  const float* x    = (const float*)d_in[0];
  const int*   mask = (const int*)d_in[1];
  float*       out  = (float*)d_out;
  hyp_fold_kernel<<<dim3(kB), dim3(32), 0, stream>>>(x, mask, out);
}